// BahdanauAttention_51204600103644
// MI455X (gfx1250) — compile-verified
//
#include <hip/hip_runtime.h>
#include <math.h>

// Bahdanau attention, MI455X (gfx1250, wave32).
// keys (256MB) read twice -> ~22us HBM floor @23.3TB/s; 137 GFLOP fp32 GEMM is
// sub-us on WMMA => memory bound => stay fp32 (V_WMMA_F32_16X16X4_F32).
// M-tile = 32 rows/WG halves Ua_w L2 re-read traffic (16GB -> 8GB) and gives
// 2 WMMAs per B-fragment load. keys->LDS staging uses the CDNA5 async path
// (GLOBAL_LOAD_ASYNC_TO_LDS_B128 + s_wait_asynccnt), no VGPR round-trip.

typedef __attribute__((ext_vector_type(2))) float v2f;
typedef __attribute__((ext_vector_type(8))) float v8f;

constexpr int BB = 32;
constexpr int SS = 2048;
constexpr int HH = 1024;
constexpr int LDSW = HH + 4;   // +4 floats: rows 16B-aligned (async b128) and
                               // A-frag ds_load_b64 bank-conflict-free.
constexpr int TM = 32;         // M rows per workgroup (2 x 16-row WMMA tiles)

// ---------------------------------------------------------------------------
// Kernel 1: q_proj[b,n] = dot(query[b,:], Wa_w[n,:]) + Wa_b[n]   (B x H, tiny)
// ---------------------------------------------------------------------------
__global__ void __launch_bounds__(256) qproj_kernel(
    const float* __restrict__ query, const float* __restrict__ Wa_w,
    const float* __restrict__ Wa_b, float* __restrict__ qproj) {
  int t = blockIdx.x * 256 + threadIdx.x;        // 0 .. B*H-1
  int b = t >> 10;
  int n = t & (HH - 1);
  const float* q = query + (size_t)b * HH;
  const float* w = Wa_w + (size_t)n * HH;
  float acc = Wa_b[n];
#pragma unroll 4
  for (int k = 0; k < HH; ++k) acc = fmaf(q[k], w[k], acc);
  qproj[t] = acc;
}

// ---------------------------------------------------------------------------
// Kernel 2: fused  k_proj = keys @ Ua_w^T  (WMMA fp32)  -> tanh -> dot(Va)
// Workgroup = 32 rows of (b,s); 8 waves x 128-col slices cover H=1024.
// ---------------------------------------------------------------------------
__global__ void __launch_bounds__(256) scores_kernel(
    const float* __restrict__ keys, const float* __restrict__ Ua_w,
    const float* __restrict__ Ua_b, const float* __restrict__ Va_w,
    const float* __restrict__ qproj, float* __restrict__ scores) {
  extern __shared__ float lds[];                 // TM*LDSW keys tile + TM accum
  float* slds = lds + TM * LDSW;

  const int tid  = threadIdx.x;
  const int row0 = blockIdx.x * TM;              // flat row = b*S + s
  const int b    = row0 / SS;                    // tile never crosses batches
  const int s0   = row0 - b * SS;

  // Stage TM x 1024 keys tile into LDS via async b128 copies (ASYNCcnt path).
  // Each chunk = 16B. LDS dest given as byte offset (dynamic LDS base = 0).
  {
    const float* src = keys + (size_t)row0 * HH;
    for (int i = tid; i < TM * (HH / 4); i += 256) {
      int r  = i >> 8;                           // row (256 chunks per row)
      int cc = i & 255;                          // 16B chunk within row
      unsigned lo = (unsigned)((r * LDSW + cc * 4) * 4);
      unsigned long long ga =
          (unsigned long long)(uintptr_t)(src + (size_t)r * HH + cc * 4);
      asm volatile("global_load_async_to_lds_b128 %0, %1, off"
                   :: "v"(lo), "v"(ga) : "memory");
    }
  }
  if (tid < TM) slds[tid] = 0.0f;
#if __has_builtin(__builtin_amdgcn_s_wait_asynccnt)
  __builtin_amdgcn_s_wait_asynccnt(0);
#else
  asm volatile("s_wait_asynccnt 0x0" ::: "memory");
#endif
  __syncthreads();

  const int wave = tid >> 5;                     // 0..7  -> 128-col slice
  const int lane = tid & 31;
  const int l15  = lane & 15;
  const int hh   = lane >> 4;                    // lane half: K split
  const int n0   = wave * 128;

  v8f acc0[8], acc1[8];
#pragma unroll
  for (int nt = 0; nt < 8; ++nt) {
    acc0[nt] = (v8f){0, 0, 0, 0, 0, 0, 0, 0};
    acc1[nt] = (v8f){0, 0, 0, 0, 0, 0, 0, 0};
  }

  // A-frag (16x4 f32): lane l: M=l&15; VGPR v holds K = k0 + v + 2*(l>>4)
  const float* lA0 = lds + l15 * LDSW + 2 * hh;          // M-tile 0
  const float* lA1 = lds + (16 + l15) * LDSW + 2 * hh;   // M-tile 1
  // B-frag (4x16 f32): lane l: N=l&15; VGPR v holds K = k0 + v + 2*(l>>4)
  // B[k,n] = Ua_w[n*H + k]  (torch Linear transposes)
  const float* gB = Ua_w + 2 * hh;

  for (int k0 = 0; k0 < HH; k0 += 4) {
    v2f a0, a1;
    a0.x = lA0[k0]; a0.y = lA0[k0 + 1];
    a1.x = lA1[k0]; a1.y = lA1[k0 + 1];
#pragma unroll
    for (int nt = 0; nt < 8; ++nt) {
      const float* bp = gB + (size_t)(n0 + nt * 16 + l15) * HH + k0;
      v2f bf;
      bf.x = bp[0];
      bf.y = bp[1];
      acc0[nt] = __builtin_amdgcn_wmma_f32_16x16x4_f32(
          false, a0, false, bf, (short)0, acc0[nt], false, false);
      acc1[nt] = __builtin_amdgcn_wmma_f32_16x16x4_f32(
          false, a1, false, bf, (short)0, acc1[nt], false, false);
    }
  }

  // Epilogue: energy = tanh(k_proj + q_proj + Ua_b); partial score = Va . energy
  // C/D layout: VGPR vg -> row M = mt*16 + vg + 8*hh ; col N = n0 + nt*16 + l15
  float p0[8] = {0, 0, 0, 0, 0, 0, 0, 0};
  float p1[8] = {0, 0, 0, 0, 0, 0, 0, 0};
#pragma unroll
  for (int nt = 0; nt < 8; ++nt) {
    int n = n0 + nt * 16 + l15;
    float qb = qproj[(size_t)b * HH + n] + Ua_b[n];
    float va = Va_w[n];
#pragma unroll
    for (int vg = 0; vg < 8; ++vg) {
      p0[vg] = fmaf(va, tanhf(acc0[nt][vg] + qb), p0[vg]);
      p1[vg] = fmaf(va, tanhf(acc1[nt][vg] + qb), p1[vg]);
    }
  }
#pragma unroll
  for (int vg = 0; vg < 8; ++vg) {
    atomicAdd(&slds[vg + 8 * hh], p0[vg]);       // ds_add_f32
    atomicAdd(&slds[16 + vg + 8 * hh], p1[vg]);
  }
  __syncthreads();

  // Va_b dropped: softmax is shift-invariant, weights/context unaffected.
  if (tid < TM) scores[(size_t)b * SS + s0 + tid] = slds[tid];
}

// ---------------------------------------------------------------------------
// Kernel 3: softmax over S per batch -> weights (an output)
// ---------------------------------------------------------------------------
__global__ void __launch_bounds__(256) softmax_kernel(
    const float* __restrict__ scores, float* __restrict__ weights) {
  __shared__ float red[256];
  const int b = blockIdx.x, tid = threadIdx.x;
  const float* s = scores + (size_t)b * SS;

  float mx = -INFINITY;
  for (int i = tid; i < SS; i += 256) mx = fmaxf(mx, s[i]);
  red[tid] = mx;
  __syncthreads();
  for (int off = 128; off > 0; off >>= 1) {
    if (tid < off) red[tid] = fmaxf(red[tid], red[tid + off]);
    __syncthreads();
  }
  mx = red[0];
  __syncthreads();

  float sum = 0.0f;
  for (int i = tid; i < SS; i += 256) sum += expf(s[i] - mx);
  red[tid] = sum;
  __syncthreads();
  for (int off = 128; off > 0; off >>= 1) {
    if (tid < off) red[tid] += red[tid + off];
    __syncthreads();
  }
  float inv = 1.0f / red[0];

  for (int i = tid; i < SS; i += 256)
    weights[(size_t)b * SS + i] = expf(s[i] - mx) * inv;
}

// ---------------------------------------------------------------------------
// Kernel 4: context[b,h] = sum_s weights[b,s] * keys[b,s,h]
// 512 blocks: 64 h-columns x 4 S-stripes per block, deterministic LDS reduce.
// ---------------------------------------------------------------------------
__global__ void __launch_bounds__(256) context_kernel(
    const float* __restrict__ keys, const float* __restrict__ weights,
    float* __restrict__ context) {
  __shared__ float red[256];
  const int b  = blockIdx.x;
  const int h  = blockIdx.y * 64 + (threadIdx.x & 63);
  const int st = threadIdx.x >> 6;               // S-stripe 0..3
  const float* kb = keys + (size_t)b * SS * HH + h;
  const float* wb = weights + (size_t)b * SS;
  float acc = 0.0f;
#pragma unroll 4
  for (int s = st * 512; s < (st + 1) * 512; ++s)
    acc = fmaf(wb[s], kb[(size_t)s * HH], acc);
  red[threadIdx.x] = acc;
  __syncthreads();
  if (st == 0)
    context[(size_t)b * HH + h] =
        ((red[threadIdx.x] + red[threadIdx.x + 64]) +
         (red[threadIdx.x + 128] + red[threadIdx.x + 192]));
}

// ---------------------------------------------------------------------------
extern "C" void kernel_launch(void* const* d_in, const int* in_sizes, int n_in,
                              void* d_out, int out_size, void* d_ws, size_t ws_size,
                              hipStream_t stream) {
  const float* query = (const float*)d_in[0];   // (B,1,H)
  const float* keys  = (const float*)d_in[1];   // (B,S,H)
  const float* Wa_w  = (const float*)d_in[2];   // (H,H)
  const float* Wa_b  = (const float*)d_in[3];   // (H)
  const float* Ua_w  = (const float*)d_in[4];   // (H,H)
  const float* Ua_b  = (const float*)d_in[5];   // (H)
  const float* Va_w  = (const float*)d_in[6];   // (1,H)
  // d_in[7] = Va_b: cancels in softmax. d_in[8] = idx: unused.

  float* context = (float*)d_out;               // B*H floats
  float* weights = (float*)d_out + BB * HH;     // B*S floats
  float* qproj   = (float*)d_ws;                // B*H floats
  float* scores  = (float*)d_ws + BB * HH;      // B*S floats

  qproj_kernel<<<(BB * HH) / 256, 256, 0, stream>>>(query, Wa_w, Wa_b, qproj);

  // TM*LDSW tile + TM score accumulators (~128.7KB; gfx1250 WGP: 320KB LDS)
  size_t shmem = (size_t)(TM * LDSW + TM) * sizeof(float);
  scores_kernel<<<(BB * SS) / TM, 256, shmem, stream>>>(keys, Ua_w, Ua_b, Va_w,
                                                        qproj, scores);

  softmax_kernel<<<BB, 256, 0, stream>>>(scores, weights);

  context_kernel<<<dim3(BB, HH / 64), 256, 0, stream>>>(keys, weights, context);
}